// DiskLoss_58918361366737
// MI455X (gfx1250) — compile-verified
//
#include <hip/hip_runtime.h>
#include <math.h>

#define BB 4
#define NN 4096
#define HH 480
#define WW 640
#define RAD2 4.0f
#define TH_SCORE 0.1f
#define KP_PENALTY (-0.001f)

typedef __attribute__((ext_vector_type(2))) float v2f;
typedef __attribute__((ext_vector_type(8))) float v8f;

// ---------------------------------------------------------------------------
// 1) bilinear grid_sample (align_corners=True) + pixel coords + |p|^2
// ---------------------------------------------------------------------------
__global__ void disk_sample(const float* __restrict__ lp, const float* __restrict__ kp,
                            float* __restrict__ scores, float* __restrict__ kx,
                            float* __restrict__ ky, float* __restrict__ sq) {
    int gid = blockIdx.x * blockDim.x + threadIdx.x;
    if (gid >= BB * NN) return;
    int img = gid / NN;
    float px = kp[2 * gid + 0];
    float py = kp[2 * gid + 1];
    float x = (px + 1.0f) * 0.5f * (float)(WW - 1);
    float y = (py + 1.0f) * 0.5f * (float)(HH - 1);
    float x0 = floorf(x), y0 = floorf(y);
    float wx = x - x0, wy = y - y0;
    const float* im = lp + (size_t)img * HH * WW;

    auto g = [&](float yy, float xx) -> float {
        int yi = (int)fminf(fmaxf(yy, 0.0f), (float)(HH - 1));
        int xi = (int)fminf(fmaxf(xx, 0.0f), (float)(WW - 1));
        return im[(size_t)yi * WW + xi];
    };
    float s = g(y0, x0) * (1.0f - wx) * (1.0f - wy)
            + g(y0, x0 + 1.0f) * wx * (1.0f - wy)
            + g(y0 + 1.0f, x0) * (1.0f - wx) * wy
            + g(y0 + 1.0f, x0 + 1.0f) * wx * wy;
    scores[gid] = s;
    kx[gid] = x;
    ky[gid] = y;
    sq[gid] = x * x + y * y;
}

// ---------------------------------------------------------------------------
// 2) WMMA Gram-tile mask build: one wave per (image,row-tile); loops col-tiles.
//    V_WMMA_F32_16X16X4_F32 computes the 16x16 dot-product tile (K=2 padded).
//    d2 = sq_r + sq_c - 2*dot. The 8 ballots of a tile are wave-uniform; lanes
//    0-15 select their row's 16-bit chunk via a cndmask tree and do a single
//    predicated scatter store per tile. B-operand loads are unconditional
//    (same address for both halves) with the K2/K3 zero-pad applied as a
//    register select, so the loop body has no predicated loads.
// ---------------------------------------------------------------------------
__global__ void disk_mask(const float* __restrict__ kx, const float* __restrict__ ky,
                          const float* __restrict__ sq,
                          unsigned short* __restrict__ mask16, int* __restrict__ cnt) {
    int wave = (blockIdx.x * blockDim.x + threadIdx.x) >> 5;  // 0..1023
    int lane = threadIdx.x & 31;
    int img = wave >> 8;        // 0..3
    int rt  = wave & 255;       // row tile 0..255
    int base = img * NN;
    int rowbase = rt * 16;
    int half = lane >> 4;
    int li = lane & 15;
    bool lowHalf = (half == 0);

    // A matrix 16x4 (MxK): lanes 0-15 hold K=0,1 ; lanes 16-31 hold K=2,3 (zero)
    float ax = kx[base + rowbase + li];
    float ay = ky[base + rowbase + li];
    v2f a;
    a.x = lowHalf ? ax : 0.0f;
    a.y = lowHalf ? ay : 0.0f;

    // per-lane row |p|^2 for C/D layout row M = v + 8*half
    float sr[8];
#pragma unroll
    for (int v = 0; v < 8; ++v) sr[v] = sq[base + rowbase + half * 8 + v];

    int myCnt = 0;  // lane l<16 accumulates neighbor count of row rowbase+l
    // lane l<16 writes row rowbase+l : v = l&7, high half if l>=8
    unsigned short* myRow = mask16 + (size_t)(base + rowbase + lane) * 256;

    for (int ct = 0; ct < 256; ++ct) {
        int colbase = ct * 16;
        // B matrix 4x16 (KxN): unconditional loads, zero-pad via select
        float bx = kx[base + colbase + li];
        float by = ky[base + colbase + li];
        v2f b;
        b.x = lowHalf ? bx : 0.0f;
        b.y = lowHalf ? by : 0.0f;
        float sc = sq[base + colbase + li];  // N = li in both halves

        v8f c = {};
        v8f d = __builtin_amdgcn_wmma_f32_16x16x4_f32(
            /*neg_a=*/false, a, /*neg_b=*/false, b,
            /*c_mod=*/(short)0, c, /*reuse_a=*/false, /*reuse_b=*/false);

        unsigned bals[8];
#pragma unroll
        for (int v = 0; v < 8; ++v) {
            // element: row M = v + 8*half (rows rowbase+M), col N = li
            float d2 = sr[v] + sc - 2.0f * d[v];
            bals[v] = __builtin_amdgcn_ballot_w32(d2 < RAD2);
        }
        // lane l selects ballot (l&7); high 16 bits if l&8 (rows +8)
        unsigned b01 = (lane & 1) ? bals[1] : bals[0];
        unsigned b23 = (lane & 1) ? bals[3] : bals[2];
        unsigned b45 = (lane & 1) ? bals[5] : bals[4];
        unsigned b67 = (lane & 1) ? bals[7] : bals[6];
        unsigned b03 = (lane & 2) ? b23 : b01;
        unsigned b47 = (lane & 2) ? b67 : b45;
        unsigned sel = (lane & 4) ? b47 : b03;
        unsigned bits = (lane & 8) ? (sel >> 16) : (sel & 0xFFFFu);
        myCnt += __popc(bits);
        if (lane < 16) myRow[ct] = (unsigned short)bits;
    }
    if (lane < 16) cnt[base + rowbase + lane] = myCnt;
}

// ---------------------------------------------------------------------------
// 3) faithful sequential NMS: one 1024-thread workgroup per image, state in LDS
//    wave32 shfl_xor argmax (first-occurrence tie-break) + 32-partial LDS pass;
//    next row's mask bits prefetched to overlap the cond broadcast.
// ---------------------------------------------------------------------------
__global__ void __launch_bounds__(1024)
disk_nms(const float* __restrict__ scores, const int* __restrict__ cnt,
         const unsigned int* __restrict__ mask32, int* __restrict__ vcOut) {
    __shared__ float sS[NN];
    __shared__ int sVC[NN];
    __shared__ unsigned char sNeed[NN];
    __shared__ float rS[32];
    __shared__ int rI[32];
    __shared__ int sCond;
    __shared__ int sJ;

    int img = blockIdx.x;
    int tid = threadIdx.x;
    int lane = tid & 31;
    int wid = tid >> 5;
    const float* sc = scores + img * NN;
    const int* c0 = cnt + img * NN;
    const unsigned* mimg = mask32 + (size_t)img * NN * 128;

    for (int k = tid; k < NN; k += 1024) {
        sS[k] = sc[k];
        int c = c0[k];
        sVC[k] = c;
        sNeed[k] = (c > 1) ? 1 : 0;  // indices_need_nms, fixed up-front
    }
    __syncthreads();

    for (int i = 0; i < NN; ++i) {
        // overlap: pull next row's 512B of mask bits toward the WGP
        if (tid < 4 && i + 1 < NN)
            __builtin_prefetch((const char*)(mimg + (size_t)(i + 1) * 128) + tid * 128, 0, 0);
        if (tid == 0) sCond = (sNeed[i] && sVC[i] > 0) ? 1 : 0;
        __syncthreads();
        if (sCond) {
            const unsigned* row = mimg + (size_t)i * 128;
            float bs = -__builtin_inff();
            int bi = NN;
#pragma unroll
            for (int t = 0; t < 4; ++t) {
                int k = tid + t * 1024;
                unsigned w = row[k >> 5];
                if ((w >> (k & 31)) & 1u) {
                    float s = sS[k];
                    if (s > bs || (s == bs && k < bi)) { bs = s; bi = k; }
                }
            }
            // wave32 reduction, tie-break lowest index (== jnp.argmax semantics)
#pragma unroll
            for (int m = 16; m > 0; m >>= 1) {
                float s2 = __shfl_xor(bs, m, 32);
                int i2 = __shfl_xor(bi, m, 32);
                if (s2 > bs || (s2 == bs && i2 < bi)) { bs = s2; bi = i2; }
            }
            if (lane == 0) { rS[wid] = bs; rI[wid] = bi; }
            __syncthreads();
            if (wid == 0) {
                bs = rS[lane]; bi = rI[lane];
#pragma unroll
                for (int m = 16; m > 0; m >>= 1) {
                    float s2 = __shfl_xor(bs, m, 32);
                    int i2 = __shfl_xor(bi, m, 32);
                    if (s2 > bs || (s2 == bs && i2 < bi)) { bs = s2; bi = i2; }
                }
                if (lane == 0) sJ = bi;
            }
            __syncthreads();
            int j = sJ;
#pragma unroll
            for (int t = 0; t < 4; ++t) {
                int k = tid + t * 1024;
                unsigned w = row[k >> 5];
                if (((w >> (k & 31)) & 1u) && k != j) sVC[k] = 0;
            }
            __syncthreads();
        }
        __syncthreads();
    }
    for (int k = tid; k < NN; k += 1024) vcOut[img * NN + k] = sVC[k];
}

// ---------------------------------------------------------------------------
// 4) per-stream reductions into acc[4]: peakySum, peakyCnt, scoreSum, validCnt
// ---------------------------------------------------------------------------
__global__ void disk_reduce(const float* __restrict__ scores, const float* __restrict__ disp,
                            const int* __restrict__ vc, float* __restrict__ acc) {
    __shared__ float r0[256], r1[256], r2[256], r3[256];
    int tid = threadIdx.x;
    int gid = blockIdx.x * blockDim.x + tid;
    float ps = 0.0f, pc = 0.0f, ss = 0.0f, vn = 0.0f;
    if (gid < BB * NN) {
        bool valid = vc[gid] > 0;
        float s = scores[gid];
        bool peaky = valid && (s > TH_SCORE);
        ps = peaky ? disp[gid] : 0.0f;
        pc = peaky ? 1.0f : 0.0f;
        ss = valid ? s : 0.0f;
        vn = valid ? 1.0f : 0.0f;
    }
    r0[tid] = ps; r1[tid] = pc; r2[tid] = ss; r3[tid] = vn;
    __syncthreads();
    for (int off = 128; off > 0; off >>= 1) {
        if (tid < off) {
            r0[tid] += r0[tid + off];
            r1[tid] += r1[tid + off];
            r2[tid] += r2[tid + off];
            r3[tid] += r3[tid + off];
        }
        __syncthreads();
    }
    if (tid == 0) {
        atomicAdd(&acc[0], r0[0]);
        atomicAdd(&acc[1], r1[0]);
        atomicAdd(&acc[2], r2[0]);
        atomicAdd(&acc[3], r3[0]);
    }
}

__global__ void disk_zero(float* __restrict__ acc) {
    if (threadIdx.x < 8) acc[threadIdx.x] = 0.0f;
}

__global__ void disk_final(const float* __restrict__ acc, float* __restrict__ out) {
    float l0 = acc[0] / fmaxf(acc[1], 1.0f) + KP_PENALTY * (acc[2] / fmaxf(acc[3], 1.0f));
    float l1 = acc[4] / fmaxf(acc[5], 1.0f) + KP_PENALTY * (acc[6] / fmaxf(acc[7], 1.0f));
    out[0] = l0 + l1;
}

// ---------------------------------------------------------------------------
extern "C" void kernel_launch(void* const* d_in, const int* in_sizes, int n_in,
                              void* d_out, int out_size, void* d_ws, size_t ws_size,
                              hipStream_t stream) {
    (void)in_sizes; (void)n_in; (void)out_size; (void)ws_size;

    // workspace layout (reused by both streams; ~8.8 MB)
    char* ws = (char*)d_ws;
    const size_t maskBytes = (size_t)BB * NN * (NN / 8);  // 8 MiB of packed bits
    unsigned short* mask16 = (unsigned short*)ws;
    unsigned int*   mask32 = (unsigned int*)ws;
    float* scores = (float*)(ws + maskBytes);
    float* kx = scores + BB * NN;
    float* ky = kx + BB * NN;
    float* sq = ky + BB * NN;
    int*   cnt = (int*)(sq + BB * NN);
    int*   vc  = cnt + BB * NN;
    float* acc = (float*)(vc + BB * NN);  // 8 floats: 4 per stream

    disk_zero<<<1, 32, 0, stream>>>(acc);

    for (int s = 0; s < 2; ++s) {
        const float* lp   = (const float*)d_in[0 + s];  // [B,1,H,W]
        const float* kp   = (const float*)d_in[2 + s];  // [B,N,2]
        const float* disp = (const float*)d_in[4 + s];  // [B,N]

        disk_sample<<<(BB * NN + 255) / 256, 256, 0, stream>>>(lp, kp, scores, kx, ky, sq);
        // 1024 waves total = B images * 256 row tiles; 8 waves per block
        disk_mask<<<128, 256, 0, stream>>>(kx, ky, sq, mask16, cnt);
        disk_nms<<<BB, 1024, 0, stream>>>(scores, cnt, mask32, vc);
        disk_reduce<<<(BB * NN + 255) / 256, 256, 0, stream>>>(scores, disp, vc, acc + 4 * s);
    }
    disk_final<<<1, 1, 0, stream>>>(acc, (float*)d_out);
}